// Model_36721970381508
// MI455X (gfx1250) — compile-verified
//
#include <hip/hip_runtime.h>
#include <math.h>

typedef __attribute__((ext_vector_type(2))) float v2f;
typedef __attribute__((ext_vector_type(8))) float v8f;

#define C_DIM 256
#define H_DIM 64
#define W_DIM 64
#define HW    4096            // H*W
#define K_DIM 64
#define OW    256             // 4*W
#define NTHR  256

// d_out layout (floats, tuple order): joints | valid | new_memory | min_dist
#define OUT_JOINTS   0
#define OUT_VALID    128
#define OUT_NEWMEM   192
#define OUT_MINDIST  16576

// ---------------- norms ----------------
__global__ void qnorm_kernel(const float* __restrict__ q, float* __restrict__ qnorm) {
    int p = blockIdx.x * blockDim.x + threadIdx.x;
    if (p >= HW) return;
    float s = 0.0f;
#pragma unroll 8
    for (int c = 0; c < C_DIM; ++c) { float v = q[c * HW + p]; s = fmaf(v, v, s); }
    qnorm[p] = s;
}

__global__ void mnorm_kernel(const float* __restrict__ mem, float* __restrict__ mnorm) {
    int k = threadIdx.x;
    if (k >= K_DIM) return;
    float s = 0.0f;
#pragma unroll 8
    for (int c = 0; c < C_DIM; ++c) { float v = mem[k * C_DIM + c]; s = fmaf(v, v, s); }
    mnorm[k] = s;
}

// ---------------- WMMA distance GEMM ----------------
// dist[n, p] = sqrt(max(||q_p||^2 - 2 * (q_p . m_n) + ||m_n||^2, 0))
// One wave per 16(pixel) x 16(memrow) tile; f32 WMMA 16x16x4, K=256 in 64 steps.
__global__ __launch_bounds__(32)
void dist_gemm_kernel(const float* __restrict__ q, const float* __restrict__ mem,
                      const float* __restrict__ qnorm, const float* __restrict__ mnorm,
                      float* __restrict__ dist) {
    const int tile_p = blockIdx.x;        // 0..255
    const int tile_n = blockIdx.y;        // 0..3
    const int lane   = threadIdx.x;       // 0..31 (wave32)
    const int half   = lane >> 4;         // 0 or 1
    const int l      = lane & 15;

    const int p = tile_p * 16 + l;        // A row (M) this lane loads
    const int n = tile_n * 16 + l;        // B col (N) this lane loads

    const float* qa = q + p;              // + c*HW
    const float* mb = mem + n * C_DIM;    // + c

    v8f acc0 = {}; v8f acc1 = {};
#pragma unroll 4
    for (int c = 0; c < C_DIM; c += 8) {
        // chain 0: K-slab [c, c+4)
        {
            const int ca = c + 2 * half;
            v2f a; a.x = qa[(ca    ) * HW];
                   a.y = qa[(ca + 1) * HW];
            v2f b = *(const v2f*)(mb + ca);       // 8B aligned (ca even)
            acc0 = __builtin_amdgcn_wmma_f32_16x16x4_f32(
                false, a, false, b, (short)0, acc0, false, false);
        }
        // chain 1: K-slab [c+4, c+8)
        {
            const int cb = c + 4 + 2 * half;
            v2f a; a.x = qa[(cb    ) * HW];
                   a.y = qa[(cb + 1) * HW];
            v2f b = *(const v2f*)(mb + cb);
            acc1 = __builtin_amdgcn_wmma_f32_16x16x4_f32(
                false, a, false, b, (short)0, acc1, false, false);
        }
    }

    // Epilogue: lane (half,l), VGPR v holds D[M = v + 8*half][N = l]
    const float mn = mnorm[tile_n * 16 + l];
#pragma unroll
    for (int v = 0; v < 8; ++v) {
        const int pm = tile_p * 16 + v + 8 * half;
        const float dot = acc0[v] + acc1[v];
        const float d2  = qnorm[pm] - 2.0f * dot + mn;
        dist[(tile_n * 16 + l) * HW + pm] = sqrtf(fmaxf(d2, 0.0f));
    }
}

// ---------------- per-k min/argmin, valid, new_memory ----------------
__global__ __launch_bounds__(NTHR)
void min_reduce_kernel(const float* __restrict__ dist, const float* __restrict__ q,
                       const float* __restrict__ mem, float* __restrict__ out) {
    const int k = blockIdx.x;
    const int t = threadIdx.x;
    __shared__ float rval[NTHR];
    __shared__ int   ridx[NTHR];
    __shared__ int   s_p;
    __shared__ int   s_valid;

    const float* dk = dist + k * HW;
    float bv = INFINITY; int bi = 0;
#pragma unroll
    for (int j = 0; j < HW / NTHR; ++j) {
        const int p = t + j * NTHR;               // coalesced
        const float v = dk[p];
        if (v < bv || (v == bv && p < bi)) { bv = v; bi = p; }
    }
    rval[t] = bv; ridx[t] = bi;
    __syncthreads();
    for (int s = NTHR / 2; s > 0; s >>= 1) {
        if (t < s) {
            const float v2 = rval[t + s]; const int i2 = ridx[t + s];
            if (v2 < rval[t] || (v2 == rval[t] && i2 < ridx[t])) { rval[t] = v2; ridx[t] = i2; }
        }
        __syncthreads();
    }
    if (t == 0) {
        const float md = rval[0];
        const int valid = ((int)truncf(md)) <= 1;   // THRESHOLD = 1
        out[OUT_MINDIST + k] = md;
        out[OUT_VALID + k]   = valid ? 1.0f : 0.0f;
        s_p = ridx[0]; s_valid = valid;
    }
    __syncthreads();
    // new_memory row k: q[:, mv, mh] if valid else memory[k, :]; thread t = channel
    const int p = s_p;
    const float nk = s_valid ? q[t * HW + p] : mem[k * C_DIM + t];
    out[OUT_NEWMEM + k * C_DIM + t] = nk;
}

// ---------------- x4 bilinear upsample + argmin -> joints ----------------
__global__ __launch_bounds__(NTHR)
void up_argmin_kernel(const float* __restrict__ dist, float* __restrict__ out) {
    const int k = blockIdx.x;
    const int t = threadIdx.x;                 // output row oy = t
    __shared__ float sd[HW];                   // 16 KB: dist[k] field
    __shared__ float rval[NTHR];
    __shared__ int   ridx[NTHR];

    const float* dk = dist + k * HW;
    for (int j = t; j < HW; j += NTHR) sd[j] = dk[j];
    __syncthreads();

    // source-y for this output row (exact in f32: steps of 0.25)
    const float cy = fmaxf((t + 0.5f) * 0.25f - 0.5f, 0.0f);
    int y0 = (int)cy; y0 = y0 > H_DIM - 1 ? H_DIM - 1 : y0;
    const float fy = cy - (float)y0;
    const int y1 = (y0 + 1 > H_DIM - 1) ? H_DIM - 1 : y0 + 1;
    const float* r0 = sd + y0 * W_DIM;
    const float* r1 = sd + y1 * W_DIM;
    const float w0y = 1.0f - fy, w1y = fy;

    float bv = INFINITY; int bi = 0;
    for (int ox = 0; ox < OW; ++ox) {
        const float cx = fmaxf((ox + 0.5f) * 0.25f - 0.5f, 0.0f);
        int x0 = (int)cx; x0 = x0 > W_DIM - 1 ? W_DIM - 1 : x0;
        const float fx = cx - (float)x0;
        const int x1 = (x0 + 1 > W_DIM - 1) ? W_DIM - 1 : x0 + 1;
        const float w00 = w0y * (1.0f - fx), w01 = w0y * fx;
        const float w10 = w1y * (1.0f - fx), w11 = w1y * fx;
        const float v = r0[x0] * w00 + r0[x1] * w01 + r1[x0] * w10 + r1[x1] * w11;
        if (v < bv) { bv = v; bi = t * OW + ox; }   // ascending -> first tie kept
    }
    rval[t] = bv; ridx[t] = bi;
    __syncthreads();
    for (int s = NTHR / 2; s > 0; s >>= 1) {
        if (t < s) {
            const float v2 = rval[t + s]; const int i2 = ridx[t + s];
            if (v2 < rval[t] || (v2 == rval[t] && i2 < ridx[t])) { rval[t] = v2; ridx[t] = i2; }
        }
        __syncthreads();
    }
    if (t == 0) {
        const int sidx = ridx[0];
        const int sv = sidx / OW, sh = sidx % OW;
        const bool valid = out[OUT_VALID + k] != 0.0f;   // written by min_reduce_kernel
        out[OUT_JOINTS + 2 * k + 0] = valid ? (float)sv : -1.0f;
        out[OUT_JOINTS + 2 * k + 1] = valid ? (float)sh : -1.0f;
    }
}

extern "C" void kernel_launch(void* const* d_in, const int* in_sizes, int n_in,
                              void* d_out, int out_size, void* d_ws, size_t ws_size,
                              hipStream_t stream) {
    const float* keymap = (const float*)d_in[0];   // [1,256,64,64] -> q
    const float* memory = (const float*)d_in[1];   // [64,256]
    float* out = (float*)d_out;

    float* ws    = (float*)d_ws;
    float* dist  = ws;                       // 64*4096 = 262144 floats (1 MB)
    float* qnorm = ws + (K_DIM * HW);        // 4096 floats
    float* mnorm = qnorm + HW;               // 64 floats

    qnorm_kernel<<<HW / NTHR, NTHR, 0, stream>>>(keymap, qnorm);
    mnorm_kernel<<<1, K_DIM, 0, stream>>>(memory, mnorm);

    dim3 gemm_grid(HW / 16, K_DIM / 16);     // 256 x 4 wave-tiles
    dist_gemm_kernel<<<gemm_grid, 32, 0, stream>>>(keymap, memory, qnorm, mnorm, dist);

    min_reduce_kernel<<<K_DIM, NTHR, 0, stream>>>(dist, keymap, memory, out);
    up_argmin_kernel<<<K_DIM, NTHR, 0, stream>>>(dist, out);
}